// GCN_53051436040231
// MI455X (gfx1250) — compile-verified
//
#include <hip/hip_runtime.h>

typedef __attribute__((ext_vector_type(2))) float v2f;
typedef __attribute__((ext_vector_type(8))) float v8f;

// ---------------- degree / normalization ----------------

__global__ __launch_bounds__(256) void deg_init_kernel(float* __restrict__ deg, int N) {
    int i = blockIdx.x * 256 + threadIdx.x;
    if (i < N) deg[i] = 1.0f;  // self-loop contributes 1 to in-degree
}

__global__ __launch_bounds__(256) void deg_accum_kernel(const int* __restrict__ dst,
                                                        float* __restrict__ deg, int E) {
    int e = blockIdx.x * 256 + threadIdx.x;
    if (e < E) atomicAdd(&deg[dst[e]], 1.0f);
}

__global__ __launch_bounds__(256) void dinv_kernel(float* __restrict__ deg, int N) {
    int i = blockIdx.x * 256 + threadIdx.x;
    if (i < N) deg[i] = rsqrtf(deg[i]);  // deg >= 1 always (self-loops)
}

// ---------------- WMMA GEMM: H[N,16] = X[N,K] @ W[K,16], fp32 ----------------
// One wave computes one 16x16 output tile (16 node rows, full 16-wide hidden).
// V_WMMA_F32_16X16X4_F32 operand layout (ISA 7.12.2):
//   A 16x4: lanes 0-15 = M rows; VGPR0 = K{0 | 2}, VGPR1 = K{1 | 3} split by lane-half
//   B 4x16: lane&15 = N col; VGPR0 = K{0 | 2}, VGPR1 = K{1 | 3} split by lane-half
//   C/D 16x16: VGPR r, lanes 0-15 -> M=r, lanes 16-31 -> M=8+r; N = lane&15
template <int K>
__global__ __launch_bounds__(256) void wmma_gemm_n16(const float* __restrict__ X,
                                                     const float* __restrict__ W,
                                                     float* __restrict__ H, int N) {
    const int lane = threadIdx.x & 31;
    const int wave = blockIdx.x * (256 / 32) + (threadIdx.x >> 5);
    const int m0   = wave * 16;
    if (m0 >= N) return;               // wave-uniform: EXEC stays all-1s for WMMA

    const int half = lane >> 4;        // 0: lanes 0-15, 1: lanes 16-31
    const int l    = lane & 15;
    int row = m0 + l;
    if (row > N - 1) row = N - 1;      // clamp loads for ragged tail (stores guarded)

    const float* __restrict__ ap = X + (long long)row * K + half * 2;

    v8f acc = {};
#pragma unroll
    for (int k = 0; k < K; k += 4) {
        v2f a = *(const v2f*)(ap + k);                 // global_load_b64 (8B aligned)
        v2f b;
        b.x = W[(k + half * 2) * 16 + l];
        b.y = W[(k + half * 2 + 1) * 16 + l];
        acc = __builtin_amdgcn_wmma_f32_16x16x4_f32(
            /*neg_a=*/false, a, /*neg_b=*/false, b,
            /*c_mod=*/(short)0, acc, /*reuse_a=*/false, /*reuse_b=*/false);
    }

    const int orow = m0 + half * 8;
#pragma unroll
    for (int r = 0; r < 8; ++r) {
        if (orow + r < N) H[(long long)(orow + r) * 16 + l] = acc[r];
    }
}

// ---------------- self-loop init: out = H * dinv^2 (also clears poison) ----------------

__global__ __launch_bounds__(256) void selfloop_init_kernel(const float* __restrict__ H,
                                                            const float* __restrict__ dinv,
                                                            float* __restrict__ out, int N) {
    int tid = blockIdx.x * 256 + threadIdx.x;   // over N*16
    if (tid >= N * 16) return;
    int i = tid >> 4;
    float di = dinv[i];
    out[tid] = H[tid] * di * di;
}

// ---------------- edge scatter: out[dst] += H[src] * dinv[src]*dinv[dst] ----------------
// 16 threads per edge: lanes cover the feature dim -> coalesced L2-resident traffic.

__global__ __launch_bounds__(256) void scatter_kernel(const float* __restrict__ H,
                                                      const int* __restrict__ src,
                                                      const int* __restrict__ dst,
                                                      const float* __restrict__ dinv,
                                                      float* __restrict__ out,
                                                      long long total /* = E*16 */) {
    long long tid = (long long)blockIdx.x * 256 + threadIdx.x;
    if (tid >= total) return;
    long long e = tid >> 4;
    int f = (int)(tid & 15);
    int s = src[e];
    int d = dst[e];
    float c = dinv[s] * dinv[d];
    atomicAdd(&out[(long long)d * 16 + f], H[(long long)s * 16 + f] * c);
}

// ---------------- bias + relu (in place) ----------------

__global__ __launch_bounds__(256) void bias_relu_kernel(float* __restrict__ A,
                                                        const float* __restrict__ b, int N) {
    int tid = blockIdx.x * 256 + threadIdx.x;   // over N*16
    if (tid >= N * 16) return;
    float v = A[tid] + b[tid & 15];
    A[tid] = v > 0.0f ? v : 0.0f;
}

// ---------------- classifier head: out[N,2] = H[N,16] @ Wc[16,2] + bc ----------------

__global__ __launch_bounds__(256) void classifier_kernel(const float* __restrict__ H,
                                                         const float* __restrict__ Wc,
                                                         const float* __restrict__ bc,
                                                         float* __restrict__ out, int N) {
    int i = blockIdx.x * 256 + threadIdx.x;
    if (i >= N) return;
    float a0 = bc[0], a1 = bc[1];
#pragma unroll
    for (int f = 0; f < 16; ++f) {
        float h = H[(long long)i * 16 + f];
        a0 += h * Wc[f * 2 + 0];
        a1 += h * Wc[f * 2 + 1];
    }
    out[(long long)i * 2 + 0] = a0;
    out[(long long)i * 2 + 1] = a1;
}

// ---------------- launch ----------------

extern "C" void kernel_launch(void* const* d_in, const int* in_sizes, int n_in,
                              void* d_out, int out_size, void* d_ws, size_t ws_size,
                              hipStream_t stream) {
    const float* x  = (const float*)d_in[0];   // [N,128]
    const int*   ei = (const int*)d_in[1];     // [2,E] int32
    const float* W1 = (const float*)d_in[2];   // [128,16]
    const float* b1 = (const float*)d_in[3];   // [16]
    const float* W2 = (const float*)d_in[4];   // [16,16]
    const float* b2 = (const float*)d_in[5];   // [16]
    const float* Wc = (const float*)d_in[6];   // [16,2]
    const float* bc = (const float*)d_in[7];   // [2]
    float* out = (float*)d_out;                // [N,2]

    const int N = in_sizes[0] / 128;
    const int E = in_sizes[1] / 2;
    const int* srcp = ei;
    const int* dstp = ei + E;

    // workspace layout: dinv[N] | bufA[N*16] | bufB[N*16]  (33*N floats = 13.2 MB)
    float* ws   = (float*)d_ws;
    float* dinv = ws;
    float* bufA = ws + N;
    float* bufB = bufA + (size_t)N * 16;

    const int blkN   = (N + 255) / 256;
    const int blkE   = (E + 255) / 256;
    const int blkN16 = (N * 16 + 255) / 256;
    const long long etot = (long long)E * 16;
    const int blkE16 = (int)((etot + 255) / 256);
    const int gemmBlocks = ((N + 15) / 16 + 7) / 8;   // 8 waves (tiles) per 256-thread block

    // normalization coefficients
    deg_init_kernel<<<blkN, 256, 0, stream>>>(dinv, N);
    deg_accum_kernel<<<blkE, 256, 0, stream>>>(dstp, dinv, E);
    dinv_kernel<<<blkN, 256, 0, stream>>>(dinv, N);

    // layer 1
    wmma_gemm_n16<128><<<gemmBlocks, 256, 0, stream>>>(x, W1, bufA, N);
    selfloop_init_kernel<<<blkN16, 256, 0, stream>>>(bufA, dinv, bufB, N);
    scatter_kernel<<<blkE16, 256, 0, stream>>>(bufA, srcp, dstp, dinv, bufB, etot);
    bias_relu_kernel<<<blkN16, 256, 0, stream>>>(bufB, b1, N);

    // layer 2
    wmma_gemm_n16<16><<<gemmBlocks, 256, 0, stream>>>(bufB, W2, bufA, N);
    selfloop_init_kernel<<<blkN16, 256, 0, stream>>>(bufA, dinv, bufB, N);
    scatter_kernel<<<blkE16, 256, 0, stream>>>(bufA, srcp, dstp, dinv, bufB, etot);
    bias_relu_kernel<<<blkN16, 256, 0, stream>>>(bufB, b2, N);

    // head
    classifier_kernel<<<blkN, 256, 0, stream>>>(bufB, Wc, bc, out, N);
}